// GIN_49503793053815
// MI455X (gfx1250) — compile-verified
//
#include <hip/hip_runtime.h>

#define DFEAT 64

typedef __attribute__((ext_vector_type(2))) float v2f;
typedef __attribute__((ext_vector_type(8))) float v8f;

__device__ __forceinline__ v8f wmma_f32(v2f a, v2f b, v8f c) {
    // D = A(16x4 f32) * B(4x16 f32) + C(16x16 f32)
    return __builtin_amdgcn_wmma_f32_16x16x4_f32(false, a, false, b, (short)0, c, false, false);
}

// Cooperative fill of LDS with W pre-swizzled into WMMA B-fragment order.
// lw[s*256 + lane*8 + c*2 + v] = W[(4s + 2*(lane>>4) + v)*64 + (16c + (lane&15))]
// 2048 float2 pairs, 256 threads -> 8 pairs each.
__device__ __forceinline__ void stage_W_lds(float* lw, const float* __restrict__ W) {
    for (int p = threadIdx.x; p < 2048; p += 256) {
        int s    = p >> 7;
        int rem  = p & 127;
        int lane = rem >> 2;
        int c    = rem & 3;
        int k    = 4 * s + 2 * (lane >> 4);
        int n    = 16 * c + (lane & 15);
        float2 w;
        w.x = W[(size_t)k * DFEAT + n];
        w.y = W[(size_t)(k + 1) * DFEAT + n];
        *(float2*)&lw[p * 2] = w;
    }
    __syncthreads();
}

// ---------------- utility kernels ----------------

__global__ void zero_kernel(float* __restrict__ p, int n) {
    int i = blockIdx.x * blockDim.x + threadIdx.x;
    if (i < n) p[i] = 0.0f;
}

__global__ void copy_kernel(float4* __restrict__ dst, const float4* __restrict__ src, int n4) {
    int i = blockIdx.x * blockDim.x + threadIdx.x;
    if (i < n4) dst[i] = src[i];
}

// ---------------- edge scatter-add: agg[dst] += h[src] ----------------
// 16 threads per edge, float4 each (64 floats per edge).
__global__ void edge_agg_kernel(const float* __restrict__ h,
                                const int* __restrict__ src,
                                const int* __restrict__ dst,
                                float* __restrict__ agg, int nedges) {
    long long t = (long long)blockIdx.x * blockDim.x + threadIdx.x;
    int e = (int)(t >> 4);
    int q = (int)(t & 15);
    if (e >= nedges) return;
    int s = src[e];
    int d = dst[e];
    const float4 v = *(const float4*)(h + (size_t)s * DFEAT + q * 4);
    float* p = agg + (size_t)d * DFEAT + q * 4;
    atomicAdd(p + 0, v.x);
    atomicAdd(p + 1, v.y);
    atomicAdd(p + 2, v.z);
    atomicAdd(p + 3, v.w);
}

// ---------------- GEMM: out = A(nrows x 64) @ W(64x64) + bias ----------------
// 256 threads = 8 waves; each wave computes a 16x64 row stripe via
// 4 f32 16x16 accumulator tiles, K walked in 16 steps of 4 (v_wmma_f32_16x16x4_f32).
// W fragments are staged in LDS once per block; per step each lane needs only
// two ds_load_b128 (all 4 B fragments) + one global_load_b64 (A pair).
__global__ void __launch_bounds__(256)
gemm_bias_kernel(const float* __restrict__ A,
                 const float* __restrict__ W,
                 const float* __restrict__ bias,
                 float* __restrict__ out, int nrows) {
    __shared__ float lw[4096];
    stage_W_lds(lw, W);

    const int lane = threadIdx.x & 31;
    const int wave = threadIdx.x >> 5;
    const int rowBase = (blockIdx.x * 8 + wave) * 16;

    const int mrow = lane & 15;
    const int hi   = lane >> 4;
    const int ncol = lane & 15;

    int arow = rowBase + mrow;
    if (arow >= nrows) arow = nrows - 1;  // clamp (dup rows never stored)
    const float* Ap = A + (size_t)arow * DFEAT;

    v8f acc[4];
#pragma unroll
    for (int c = 0; c < 4; ++c) {
        float bv = bias[c * 16 + ncol];
        acc[c] = (v8f){bv, bv, bv, bv, bv, bv, bv, bv};
    }

#pragma unroll
    for (int s = 0; s < 16; ++s) {
        const int k = s * 4 + hi * 2;  // this lane's K pair
        const float2 av = *(const float2*)(Ap + k);
        v2f a;
        a.x = av.x;
        a.y = av.y;
        const float4* lwp = (const float4*)&lw[s * 256 + lane * 8];
        const float4 b01 = lwp[0];  // c=0 pair, c=1 pair
        const float4 b23 = lwp[1];  // c=2 pair, c=3 pair
        v2f b;
        b.x = b01.x; b.y = b01.y; acc[0] = wmma_f32(a, b, acc[0]);
        b.x = b01.z; b.y = b01.w; acc[1] = wmma_f32(a, b, acc[1]);
        b.x = b23.x; b.y = b23.y; acc[2] = wmma_f32(a, b, acc[2]);
        b.x = b23.z; b.y = b23.w; acc[3] = wmma_f32(a, b, acc[3]);
    }

#pragma unroll
    for (int r = 0; r < 8; ++r) {
        int row = rowBase + r + 8 * hi;
        if (row < nrows) {
            float* op = out + (size_t)row * DFEAT + ncol;
#pragma unroll
            for (int c = 0; c < 4; ++c) op[c * 16] = acc[c][r];
        }
    }
}

// ---------------- per-column sum / sumsq (for BatchNorm batch stats) ----------------
__global__ void stats_kernel(const float* __restrict__ X,
                             float* __restrict__ sums,
                             float* __restrict__ sumsq, int nrows) {
    __shared__ float ls[DFEAT];
    __shared__ float lq[DFEAT];
    const int col = threadIdx.x & 63;
    const int sub = threadIdx.x >> 6;  // 4 row-lanes per column
    if (threadIdx.x < DFEAT) {
        ls[threadIdx.x] = 0.0f;
        lq[threadIdx.x] = 0.0f;
    }
    __syncthreads();
    float s = 0.0f, q = 0.0f;
    for (int row = blockIdx.x * 4 + sub; row < nrows; row += gridDim.x * 4) {
        float v = X[(size_t)row * DFEAT + col];
        s += v;
        q += v * v;
    }
    atomicAdd(&ls[col], s);   // ds_add_f32
    atomicAdd(&lq[col], q);
    __syncthreads();
    if (threadIdx.x < DFEAT) {
        atomicAdd(&sums[threadIdx.x], ls[threadIdx.x]);
        atomicAdd(&sumsq[threadIdx.x], lq[threadIdx.x]);
    }
}

__global__ void bn_finalize_kernel(const float* __restrict__ sums,
                                   const float* __restrict__ sumsq,
                                   const float* __restrict__ g,
                                   const float* __restrict__ bt,
                                   float* __restrict__ scale,
                                   float* __restrict__ shift, float invN) {
    int c = threadIdx.x;
    float mu  = sums[c] * invN;
    float var = fmaxf(sumsq[c] * invN - mu * mu, 0.0f);
    float sc  = g[c] * rsqrtf(var + 1e-5f);
    scale[c] = sc;
    shift[c] = bt[c] - mu * sc;
}

// ---------------- fused BN + ReLU + GEMM2 (+ optional outer ReLU) ----------------
__global__ void __launch_bounds__(256)
bn_relu_gemm_kernel(const float* __restrict__ X,  // pre-BN linear output
                    const float* __restrict__ scale,
                    const float* __restrict__ shift,
                    const float* __restrict__ W,
                    const float* __restrict__ bias,
                    float* __restrict__ out, int nrows, int outerRelu) {
    __shared__ float lw[4096];
    stage_W_lds(lw, W);

    const int lane = threadIdx.x & 31;
    const int wave = threadIdx.x >> 5;
    const int rowBase = (blockIdx.x * 8 + wave) * 16;

    const int mrow = lane & 15;
    const int hi   = lane >> 4;
    const int ncol = lane & 15;

    int arow = rowBase + mrow;
    if (arow >= nrows) arow = nrows - 1;
    const float* Xp = X + (size_t)arow * DFEAT;

    v8f acc[4];
#pragma unroll
    for (int c = 0; c < 4; ++c) {
        float bv = bias[c * 16 + ncol];
        acc[c] = (v8f){bv, bv, bv, bv, bv, bv, bv, bv};
    }

#pragma unroll
    for (int s = 0; s < 16; ++s) {
        const int k = s * 4 + hi * 2;
        const float2 xv = *(const float2*)(Xp + k);
        const float2 sc = *(const float2*)(scale + k);
        const float2 sh = *(const float2*)(shift + k);
        v2f a;
        a.x = fmaxf(xv.x * sc.x + sh.x, 0.0f);  // BN + ReLU on the fly
        a.y = fmaxf(xv.y * sc.y + sh.y, 0.0f);
        const float4* lwp = (const float4*)&lw[s * 256 + lane * 8];
        const float4 b01 = lwp[0];
        const float4 b23 = lwp[1];
        v2f b;
        b.x = b01.x; b.y = b01.y; acc[0] = wmma_f32(a, b, acc[0]);
        b.x = b01.z; b.y = b01.w; acc[1] = wmma_f32(a, b, acc[1]);
        b.x = b23.x; b.y = b23.y; acc[2] = wmma_f32(a, b, acc[2]);
        b.x = b23.z; b.y = b23.w; acc[3] = wmma_f32(a, b, acc[3]);
    }

#pragma unroll
    for (int r = 0; r < 8; ++r) {
        int row = rowBase + r + 8 * hi;
        if (row < nrows) {
            float* op = out + (size_t)row * DFEAT + ncol;
#pragma unroll
            for (int c = 0; c < 4; ++c) {
                float v = acc[c][r];
                if (outerRelu) v = fmaxf(v, 0.0f);
                op[c * 16] = v;
            }
        }
    }
}

// ---------------- per-graph mean pooling ----------------
__global__ void pool_kernel(const float* __restrict__ h,
                            const int* __restrict__ gid,
                            float* __restrict__ out,
                            float* __restrict__ counts, int nrows) {
    long long t = (long long)blockIdx.x * blockDim.x + threadIdx.x;
    int node = (int)(t >> 4);
    int q = (int)(t & 15);
    if (node >= nrows) return;
    int g = gid[node];
    const float4 v = *(const float4*)(h + (size_t)node * DFEAT + q * 4);
    float* p = out + (size_t)g * DFEAT + q * 4;
    atomicAdd(p + 0, v.x);
    atomicAdd(p + 1, v.y);
    atomicAdd(p + 2, v.z);
    atomicAdd(p + 3, v.w);
    if (q == 0) atomicAdd(&counts[g], 1.0f);
}

__global__ void pool_div_kernel(float* __restrict__ out, const float* __restrict__ counts, int n) {
    int i = blockIdx.x * blockDim.x + threadIdx.x;
    if (i < n) out[i] = out[i] / fmaxf(counts[i >> 6], 1.0f);
}

// ---------------- launch ----------------

extern "C" void kernel_launch(void* const* d_in, const int* in_sizes, int n_in,
                              void* d_out, int out_size, void* d_ws, size_t ws_size,
                              hipStream_t stream) {
    const float* feats = (const float*)d_in[0];
    const float* W1_0  = (const float*)d_in[1];
    const float* b1_0  = (const float*)d_in[2];
    const float* g_0   = (const float*)d_in[3];
    const float* bt_0  = (const float*)d_in[4];
    const float* W2_0  = (const float*)d_in[5];
    const float* b2_0  = (const float*)d_in[6];
    const float* W1_1  = (const float*)d_in[7];
    const float* b1_1  = (const float*)d_in[8];
    const float* g_1   = (const float*)d_in[9];
    const float* bt_1  = (const float*)d_in[10];
    const float* W2_1  = (const float*)d_in[11];
    const float* b2_1  = (const float*)d_in[12];
    const int*   src   = (const int*)d_in[13];
    const int*   dst   = (const int*)d_in[14];
    const int*   gid   = (const int*)d_in[15];

    const int n_nodes  = in_sizes[0] / DFEAT;
    const int n_edges  = in_sizes[13];
    const int n_graphs = out_size / DFEAT;
    const int NF       = n_nodes * DFEAT;
    const float invN   = 1.0f / (float)n_nodes;

    float* bufA   = (float*)d_ws;        // 25.6 MB
    float* bufB   = bufA + (size_t)NF;   // 25.6 MB
    float* sums   = bufB + (size_t)NF;   // 64
    float* sumsq  = sums + DFEAT;        // 64
    float* scale  = sumsq + DFEAT;       // 64
    float* shift  = scale + DFEAT;       // 64
    float* counts = shift + DFEAT;       // n_graphs
    float* outf   = (float*)d_out;

    const int copyBlocks  = (NF / 4 + 255) / 256;
    const int edgeBlocks  = (int)(((long long)n_edges * 16 + 255) / 256);
    const int gemmBlocks  = (n_nodes + 127) / 128;   // 8 waves * 16 rows per block
    const int statsBlocks = 512;
    const int poolBlocks  = (int)(((long long)n_nodes * 16 + 255) / 256);

    // ---------- Layer 0 ----------
    copy_kernel<<<copyBlocks, 256, 0, stream>>>((float4*)bufA, (const float4*)feats, NF / 4);
    edge_agg_kernel<<<edgeBlocks, 256, 0, stream>>>(feats, src, dst, bufA, n_edges);

    zero_kernel<<<1, 128, 0, stream>>>(sums, 2 * DFEAT);  // sums + sumsq contiguous
    gemm_bias_kernel<<<gemmBlocks, 256, 0, stream>>>(bufA, W1_0, b1_0, bufB, n_nodes);
    stats_kernel<<<statsBlocks, 256, 0, stream>>>(bufB, sums, sumsq, n_nodes);
    bn_finalize_kernel<<<1, DFEAT, 0, stream>>>(sums, sumsq, g_0, bt_0, scale, shift, invN);
    bn_relu_gemm_kernel<<<gemmBlocks, 256, 0, stream>>>(bufB, scale, shift, W2_0, b2_0,
                                                        bufA, n_nodes, /*outerRelu=*/1);
    // h1 in bufA

    // ---------- Layer 1 ----------
    copy_kernel<<<copyBlocks, 256, 0, stream>>>((float4*)bufB, (const float4*)bufA, NF / 4);
    edge_agg_kernel<<<edgeBlocks, 256, 0, stream>>>(bufA, src, dst, bufB, n_edges);
    // agg1 in bufB; h1 (bufA) now dead

    zero_kernel<<<1, 128, 0, stream>>>(sums, 2 * DFEAT);
    gemm_bias_kernel<<<gemmBlocks, 256, 0, stream>>>(bufB, W1_1, b1_1, bufA, n_nodes);
    stats_kernel<<<statsBlocks, 256, 0, stream>>>(bufA, sums, sumsq, n_nodes);
    bn_finalize_kernel<<<1, DFEAT, 0, stream>>>(sums, sumsq, g_1, bt_1, scale, shift, invN);
    bn_relu_gemm_kernel<<<gemmBlocks, 256, 0, stream>>>(bufA, scale, shift, W2_1, b2_1,
                                                        bufB, n_nodes, /*outerRelu=*/0);
    // h2 in bufB

    // ---------- Readout ----------
    zero_kernel<<<(out_size + 255) / 256, 256, 0, stream>>>(outf, out_size);
    zero_kernel<<<(n_graphs + 255) / 256, 256, 0, stream>>>(counts, n_graphs);
    pool_kernel<<<poolBlocks, 256, 0, stream>>>(bufB, gid, outf, counts, n_nodes);
    pool_div_kernel<<<(out_size + 255) / 256, 256, 0, stream>>>(outf, counts, out_size);
}